// DRCLoss_91010357002637
// MI455X (gfx1250) — compile-verified
//
#include <hip/hip_runtime.h>
#include <hip/hip_bf16.h>

typedef float v2f __attribute__((ext_vector_type(2)));
typedef float v8f __attribute__((ext_vector_type(8)));

#define NB 4096
#define ND 512
#define ROWS_PER_WG 64
#define LDS_STRIDE 516          // 512 + 4 pad -> lane16*516 gives distinct banks
#define COL_GROUPS 4
#define COL_TILES 768           // 3*NB/16
#define TILES_PER_GROUP (COL_TILES / COL_GROUPS)   // 192
#define NWAVES 8
#define TILES_PER_WAVE (TILES_PER_GROUP / NWAVES)  // 24
#define MARGIN 0.1f
#define PD_EPS 1e-6f

// ---------------------------------------------------------------------------
// Kernel 1: per-row stats. One wave per row. Also re-initializes negminBits
// (must happen every launch: graph replay does not re-poison workspace).
// ---------------------------------------------------------------------------
__global__ __launch_bounds__(256) void row_stats_kernel(
    const float* __restrict__ ts, const float* __restrict__ i1,
    const float* __restrict__ i2, float* __restrict__ norms,
    float* __restrict__ lpos1, float* __restrict__ lpos2,
    float* __restrict__ dist12, unsigned* __restrict__ negminBits)
{
    const int wid  = threadIdx.x >> 5;
    const int lane = threadIdx.x & 31;
    const int row  = blockIdx.x * NWAVES + wid;
    const float* pt = ts + (size_t)row * ND;
    const float* p1 = i1 + (size_t)row * ND;
    const float* p2 = i2 + (size_t)row * ND;

    float s0 = 0.f, s1 = 0.f, s2 = 0.f, d1 = 0.f, d2 = 0.f, dd = 0.f;
    for (int c = lane; c < ND; c += 32) {
        const float a = pt[c], b1 = p1[c], b2 = p2[c];
        s0 += a * a;  s1 += b1 * b1;  s2 += b2 * b2;
        d1 += a * b1; d2 += a * b2;
        const float df = b1 - b2 + PD_EPS;
        dd += df * df;
    }
    #pragma unroll
    for (int off = 16; off > 0; off >>= 1) {
        s0 += __shfl_xor(s0, off, 32);
        s1 += __shfl_xor(s1, off, 32);
        s2 += __shfl_xor(s2, off, 32);
        d1 += __shfl_xor(d1, off, 32);
        d2 += __shfl_xor(d2, off, 32);
        dd += __shfl_xor(dd, off, 32);
    }
    if (lane == 0) {
        norms[row]          = s0;
        norms[NB + row]     = s1;
        norms[2 * NB + row] = s2;
        float q1 = s0 + s1 - 2.f * d1;
        float q2 = s0 + s2 - 2.f * d2;
        lpos1[row]  = q1 > 0.f ? sqrtf(q1) : 0.f;
        lpos2[row]  = q2 > 0.f ? sqrtf(q2) : 0.f;
        dist12[row] = sqrtf(dd);
        negminBits[row] = 0x7F800000u;   // +inf (float bits; all dists >= 0)
    }
}

// ---------------------------------------------------------------------------
// Kernel 2: WMMA distance-min. Grid = (NB/64 row groups) x COL_GROUPS.
// A tile (64 ts rows x 512) in LDS, B tiles streamed from global (L2-resident).
// fp32 WMMA 16x16x4; 4 row-subtiles share each B fragment.
// ---------------------------------------------------------------------------
__global__ __launch_bounds__(256) void dist_min_kernel(
    const float* __restrict__ ts, const float* __restrict__ i1,
    const float* __restrict__ i2, const float* __restrict__ norms,
    unsigned* __restrict__ negminBits)
{
    __shared__ float As[ROWS_PER_WG * LDS_STRIDE];
    __shared__ float rowN[ROWS_PER_WG];

    const int tid    = threadIdx.x;
    const int wave   = tid >> 5;
    const int lane   = tid & 31;
    const int lane16 = lane & 15;
    const int h      = lane >> 4;

    const int rowGrp  = blockIdx.x >> 2;            // / COL_GROUPS
    const int colGrp  = blockIdx.x & (COL_GROUPS - 1);
    const int rowBase = rowGrp * ROWS_PER_WG;

    // Stage A tile: 64 rows x 512 floats, float4 copies, padded LDS rows.
    for (int s = tid; s < ROWS_PER_WG * (ND / 4); s += 256) {
        const int r  = s >> 7;        // / 128 float4 per row
        const int c4 = s & 127;
        const float4 v = ((const float4*)(ts + (size_t)(rowBase + r) * ND))[c4];
        *(float4*)&As[r * LDS_STRIDE + c4 * 4] = v;
    }
    if (tid < ROWS_PER_WG) rowN[tid] = norms[rowBase + tid];
    __syncthreads();

    // Row norms for the 32 (t,v) elements this lane-half owns.
    float rn[4][8];
    #pragma unroll
    for (int t = 0; t < 4; ++t)
        #pragma unroll
        for (int v = 0; v < 8; ++v)
            rn[t][v] = rowN[t * 16 + v + 8 * h];

    float runMin[4][8];
    #pragma unroll
    for (int t = 0; t < 4; ++t)
        #pragma unroll
        for (int v = 0; v < 8; ++v)
            runMin[t][v] = __builtin_inff();

    const float* aptr = &As[lane16 * LDS_STRIDE + 2 * h];

    for (int s = 0; s < TILES_PER_WAVE; ++s) {
        const int jt = colGrp * TILES_PER_GROUP + s * NWAVES + wave;
        const int j0 = jt * 16;
        const float* src; int lc;
        if (j0 < NB)          { src = ts; lc = j0; }
        else if (j0 < 2 * NB) { src = i1; lc = j0 - NB; }
        else                  { src = i2; lc = j0 - 2 * NB; }
        const float* bptr = src + (size_t)(lc + lane16) * ND + 2 * h;

        // Prefetch next tile's B stream (global_prefetch_b8).
        if (s + 1 < TILES_PER_WAVE) {
            const int jn = colGrp * TILES_PER_GROUP + (s + 1) * NWAVES + wave;
            const int jn0 = jn * 16;
            const float* srcn; int lcn;
            if (jn0 < NB)          { srcn = ts; lcn = jn0; }
            else if (jn0 < 2 * NB) { srcn = i1; lcn = jn0 - NB; }
            else                   { srcn = i2; lcn = jn0 - 2 * NB; }
            __builtin_prefetch(srcn + (size_t)(lcn + lane16) * ND, 0, 1);
        }

        v8f acc[4];
        #pragma unroll
        for (int t = 0; t < 4; ++t) acc[t] = (v8f){0.f,0.f,0.f,0.f,0.f,0.f,0.f,0.f};

        #pragma unroll 4
        for (int k = 0; k < ND; k += 4) {
            const v2f bf = *(const v2f*)(bptr + k);
            const v2f a0 = *(const v2f*)(aptr + k);
            const v2f a1 = *(const v2f*)(aptr + 16 * LDS_STRIDE + k);
            const v2f a2 = *(const v2f*)(aptr + 32 * LDS_STRIDE + k);
            const v2f a3 = *(const v2f*)(aptr + 48 * LDS_STRIDE + k);
            acc[0] = __builtin_amdgcn_wmma_f32_16x16x4_f32(false, a0, false, bf, (short)0, acc[0], false, false);
            acc[1] = __builtin_amdgcn_wmma_f32_16x16x4_f32(false, a1, false, bf, (short)0, acc[1], false, false);
            acc[2] = __builtin_amdgcn_wmma_f32_16x16x4_f32(false, a2, false, bf, (short)0, acc[2], false, false);
            acc[3] = __builtin_amdgcn_wmma_f32_16x16x4_f32(false, a3, false, bf, (short)0, acc[3], false, false);
        }

        const float cn = norms[j0 + lane16];
        const int   j  = j0 + lane16;
        #pragma unroll
        for (int t = 0; t < 4; ++t) {
            #pragma unroll
            for (int v = 0; v < 8; ++v) {
                const int i = rowBase + t * 16 + v + 8 * h;
                const float sq = rn[t][v] + cn - 2.f * acc[t][v];
                float dist = sq > 0.f ? sqrtf(sq) : 0.f;
                const bool same = (j == i) || (j == i + NB) || (j == i + 2 * NB);
                dist = same ? __builtin_inff() : dist;
                runMin[t][v] = fminf(runMin[t][v], dist);
            }
        }
    }

    // Reduce each row's 16 column-lanes, then combine across waves / WGs.
    #pragma unroll
    for (int t = 0; t < 4; ++t) {
        #pragma unroll
        for (int v = 0; v < 8; ++v) {
            float m = runMin[t][v];
            m = fminf(m, __shfl_xor(m, 8, 32));
            m = fminf(m, __shfl_xor(m, 4, 32));
            m = fminf(m, __shfl_xor(m, 2, 32));
            m = fminf(m, __shfl_xor(m, 1, 32));
            if (lane16 == 0) {
                const int row = rowBase + t * 16 + v + 8 * h;
                atomicMin(&negminBits[row], __float_as_uint(m));
            }
        }
    }
}

// ---------------------------------------------------------------------------
// Kernel 3: triplet formula + mean. Single block, deterministic.
// ---------------------------------------------------------------------------
__global__ __launch_bounds__(256) void finalize_kernel(
    const float* __restrict__ lpos1, const float* __restrict__ lpos2,
    const float* __restrict__ dist12, const unsigned* __restrict__ negminBits,
    float* __restrict__ out)
{
    __shared__ float sacc[NWAVES];
    float s = 0.f;
    for (int i = threadIdx.x; i < NB; i += 256) {
        const float p1 = lpos1[i], p2 = lpos2[i];
        const float pos = p1 + p2 + dist12[i];
        const float neg = __uint_as_float(negminBits[i]);
        float tr = pos - neg + MARGIN;
        tr = tr > 0.f ? tr : 0.f;
        s += tr + fmaxf(p1, p2);
    }
    #pragma unroll
    for (int off = 16; off > 0; off >>= 1) s += __shfl_xor(s, off, 32);
    if ((threadIdx.x & 31) == 0) sacc[threadIdx.x >> 5] = s;
    __syncthreads();
    if (threadIdx.x == 0) {
        float tot = 0.f;
        for (int w = 0; w < NWAVES; ++w) tot += sacc[w];
        out[0] = tot / (float)NB;
    }
}

// ---------------------------------------------------------------------------
extern "C" void kernel_launch(void* const* d_in, const int* in_sizes, int n_in,
                              void* d_out, int out_size, void* d_ws, size_t ws_size,
                              hipStream_t stream)
{
    const float* ts = (const float*)d_in[0];
    const float* i1 = (const float*)d_in[1];
    const float* i2 = (const float*)d_in[2];

    float* ws       = (float*)d_ws;
    float* norms    = ws;                  // 3*NB
    float* lpos1    = ws + 3 * NB;         // NB
    float* lpos2    = lpos1 + NB;          // NB
    float* dist12   = lpos2 + NB;          // NB
    unsigned* negmin = (unsigned*)(dist12 + NB); // NB  (total 28672 floats)

    row_stats_kernel<<<NB / NWAVES, 256, 0, stream>>>(ts, i1, i2, norms,
                                                      lpos1, lpos2, dist12, negmin);
    dist_min_kernel<<<(NB / ROWS_PER_WG) * COL_GROUPS, 256, 0, stream>>>(
        ts, i1, i2, norms, negmin);
    finalize_kernel<<<1, 256, 0, stream>>>(lpos1, lpos2, dist12, negmin,
                                           (float*)d_out);
}